// YeoJohnson_62053687493093
// MI455X (gfx1250) — compile-verified
//
#include <hip/hip_runtime.h>
#include <cstdint>
#include <cstddef>

// Yeo-Johnson power transform, tuned for MI455X (gfx1250, wave32):
//   - 512 MB streaming traffic -> memory bound (~22us floor @ 23.3 TB/s)
//   - one v_log_f32 + one v_exp_f32 per element (branches share log2(1+|x|))
//   - per-column reciprocals cached in registers, reused over 16 rows
//   - lambda slice staged via CDNA5 async global->LDS (ASYNCcnt) path
//   - non-temporal b128 loads/stores keep the stream out of the 192MB L2

typedef float v4f __attribute__((ext_vector_type(4)));

#define LN2F 0.69314718055994530942f
#define C_TILE 1024
#define R_TILE 16
#define TPB 256

__global__ __launch_bounds__(TPB) void yeojohnson_kernel(
    const float* __restrict__ x,
    const float* __restrict__ lmbda,
    float* __restrict__ out,
    int D)
{
    __shared__ float sh_lam[C_TILE];

    const int tid  = threadIdx.x;
    const int col0 = blockIdx.x * C_TILE;
    const int row0 = blockIdx.y * R_TILE;
    const int c    = col0 + tid * 4;

    // ---- Stage this block's 1024-column lambda slice into LDS via the
    // CDNA5 async global->LDS copy: 256 lanes x b128 = 4KB, ASYNCcnt-tracked.
    {
        unsigned lds_dst = (unsigned)(uintptr_t)(&sh_lam[tid * 4]); // LDS offset = low 32b of flat shared ptr
        unsigned voff    = (unsigned)c * 4u;                        // byte offset from lmbda base (GVS mode)
        asm volatile("global_load_async_to_lds_b128 %0, %1, %2"
                     :: "v"(lds_dst), "v"(voff), "s"(lmbda)
                     : "memory");
        asm volatile("s_wait_asynccnt 0x0" ::: "memory");
    }
    __syncthreads();

    // ---- Per-thread per-column constants, reused across R_TILE rows.
    const v4f lam = *reinterpret_cast<const v4f*>(&sh_lam[tid * 4]); // ds_load_b128
    v4f rlm, r2m, twom;
    bool islogp[4], islogn[4];
#pragma unroll
    for (int i = 0; i < 4; ++i) {
        const float lm = lam[i];
        const float tm = 2.0f - lm;
        twom[i]   = tm;
        islogp[i] = (lm == 0.0f);
        islogn[i] = (lm == 2.0f);
        rlm[i] = __builtin_amdgcn_rcpf(islogp[i] ? 1.0f : lm); // v_rcp_f32
        r2m[i] = __builtin_amdgcn_rcpf(islogn[i] ? 1.0f : tm);
    }

    // ---- 16 rows of streaming compute on this thread's 4 columns.
#pragma unroll
    for (int r = 0; r < R_TILE; ++r) {
        const size_t idx = (size_t)(row0 + r) * (size_t)D + (size_t)c;
        const v4f xv = __builtin_nontemporal_load((const v4f*)(x + idx));
        v4f ov;
#pragma unroll
        for (int i = 0; i < 4; ++i) {
            const float xi  = xv[i];
            const bool  pos = (xi >= 0.0f);
            const float ax  = __builtin_fabsf(xi);
            // shared by both sign branches: log2(1+|x|)
            const float l2  = __builtin_amdgcn_logf(1.0f + ax);   // v_log_f32
            const float e   = pos ? lam[i] : twom[i];
            const float p   = __builtin_amdgcn_exp2f(e * l2);     // v_exp_f32 -> (1+|x|)^e
            // pos: ((x+1)^lm - 1)/lm ; neg: -(((1-x)^(2-lm)) - 1)/(2-lm)
            const float pw  = pos ? (p - 1.0f) * rlm[i] : (1.0f - p) * r2m[i];
            // pos,lm==0: log1p(x) ; neg,lm==2: -log1p(-x)
            const float lg  = pos ? (l2 * LN2F) : -(l2 * LN2F);
            const bool  ul  = pos ? islogp[i] : islogn[i];
            ov[i] = ul ? lg : pw;
        }
        __builtin_nontemporal_store(ov, (v4f*)(out + idx));
    }
}

extern "C" void kernel_launch(void* const* d_in, const int* in_sizes, int n_in,
                              void* d_out, int out_size, void* d_ws, size_t ws_size,
                              hipStream_t stream) {
    (void)n_in; (void)out_size; (void)d_ws; (void)ws_size;
    const float* x     = (const float*)d_in[0];
    const float* lmbda = (const float*)d_in[1];
    float*       out   = (float*)d_out;

    const int D = in_sizes[1];              // 16384
    const int B = in_sizes[0] / D;          // 4096

    dim3 grid(D / C_TILE, B / R_TILE);      // (16, 256) = 4096 workgroups
    dim3 block(TPB);                        // 8 wave32 per block
    hipLaunchKernelGGL(yeojohnson_kernel, grid, block, 0, stream, x, lmbda, out, D);
}